// S4DConv_90623809945981
// MI455X (gfx1250) — compile-verified
//
#include <hip/hip_runtime.h>

// Problem constants (from reference): B=8, L=2048, H=512, N=64, CH=1
#define B_   8
#define L_   2048
#define H_   512
#define N_   64
#define Q_   64        // chunk length
#define NCH  32        // L_/Q_

typedef float v2f __attribute__((ext_vector_type(2)));
typedef float v8f __attribute__((ext_vector_type(8)));

// D = A(16x4) * B(4x16) + C, f32 WMMA (wave32, EXEC all ones)
__device__ __forceinline__ v8f wmma4(v2f a, v2f b, v8f c) {
    return __builtin_amdgcn_wmma_f32_16x16x4_f32(
        /*neg_a=*/false, a, /*neg_b=*/false, b,
        /*c_mod=*/(short)0, c, /*reuse_a=*/false, /*reuse_b=*/false);
}

// ---------------------------------------------------------------------------
// Kernel A: per (h,n) compute lambda = exp(dt*w), Cs = 2*C*(lambda-1)/w, and
// materialize the tables consumed by the WMMA kernels:
//   kvals[h][m]      = Re(sum_n Cs * lambda^m),  m=0..Q-1   (Toeplitz source)
//   Vre/Vim[h][n][t] = lambda^(Q-1-t)                          (state Vandermonde)
//   Gre[h][t][n]     =  Re(Cs * lambda^(t+1))
//   Gimn[h][t][n]    = -Im(Cs * lambda^(t+1))   (negated: f32 WMMA has no A-neg)
//   lamQ[h][n]       = lambda^Q
// ---------------------------------------------------------------------------
__global__ __launch_bounds__(64) void s4d_setup(
    const float* __restrict__ w_re, const float* __restrict__ w_im,
    const float* __restrict__ C_re, const float* __restrict__ C_im,
    const float* __restrict__ dt,
    float* __restrict__ Vre, float* __restrict__ Vim,
    float* __restrict__ Gre, float* __restrict__ Gimn,
    float* __restrict__ kvals, float* __restrict__ lamQ)
{
    const int h = blockIdx.x;
    const int n = threadIdx.x;
    const int hn = h * N_ + n;

    const float wr = w_re[hn], wi = w_im[hn];
    const float dtv = dt[h];
    const float dAr = wr * dtv, dAi = wi * dtv;
    const float er = expf(dAr);
    const float lr = er * cosf(dAi);
    const float li = er * sinf(dAi);
    // (lambda - 1) / w  (complex divide)
    const float numr = lr - 1.0f, numi = li;
    const float inv  = 1.0f / (wr * wr + wi * wi);
    const float qr = (numr * wr + numi * wi) * inv;
    const float qi = (numi * wr - numr * wi) * inv;
    const float Cr = C_re[hn], Ci = C_im[hn];
    const float csr = 2.0f * (Cr * qr - Ci * qi);   // factor 2 folded here
    const float csi = 2.0f * (Cr * qi + Ci * qr);

    __shared__ float kacc[Q_];
    kacc[n] = 0.0f;
    __syncthreads();

    float pr = 1.0f, pi = 0.0f;                     // lambda^m
    const size_t vbase = ((size_t)h * N_ + n) * Q_;
    const size_t gbase = (size_t)h * Q_ * N_;
    for (int m = 0; m < Q_; ++m) {
        atomicAdd(&kacc[m], csr * pr - csi * pi);   // k[m] reduction over n (LDS)
        Vre[vbase + (Q_ - 1 - m)] = pr;
        Vim[vbase + (Q_ - 1 - m)] = pi;
        const float npr = pr * lr - pi * li;
        const float npi = pr * li + pi * lr;
        pr = npr; pi = npi;                          // now lambda^(m+1)
        Gre [gbase + (size_t)m * N_ + n] =  (csr * pr - csi * pi);
        Gimn[gbase + (size_t)m * N_ + n] = -(csr * pi + csi * pr);
    }
    __syncthreads();
    kvals[hn]        = kacc[n];
    lamQ[2 * hn]     = pr;                           // lambda^Q
    lamQ[2 * hn + 1] = pi;
}

// ---------------------------------------------------------------------------
// Kernel B: one wave per (chunk j, head h).
//   out[b, j*Q+t, h] = sum_{tau<=t} k[t-tau]*u[tau]  + D[h]*u   (intra Toeplitz)
//   P[h][j][re/im][n][b] = sum_tau lambda^(Q-1-tau) * u[tau]    (state summary)
// A-fragments are fully preloaded before each 16-deep WMMA chain; the Toeplitz
// gather uses a zero-padded reversed-k LDS buffer so indices are unconditional.
// ---------------------------------------------------------------------------
__global__ __launch_bounds__(32) void s4d_chunk(
    const float* __restrict__ u, const float* __restrict__ Dvec,
    const float* __restrict__ Vre, const float* __restrict__ Vim,
    const float* __restrict__ kvals,
    float* __restrict__ P, float* __restrict__ out)
{
    const int j = blockIdx.x, h = blockIdx.y;
    const int lane = threadIdx.x;
    const int half = lane >> 4;      // lane group (K split)
    const int nl   = lane & 15;      // B-matrix column / A-matrix row
    const int l0 = j * Q_;

    // rp[i] = k[63-i] for i<64, else 0  => unconditional Toeplitz gather:
    //   T[t][tau] = k[t-tau] (tau<=t) = rp[63 - t + tau]
    __shared__ float rp[2 * Q_];
    rp[lane]      = kvals[h * Q_ + (63 - lane)];
    rp[lane + 32] = kvals[h * Q_ + (31 - lane)];
    rp[lane + 64] = 0.0f;
    rp[lane + 96] = 0.0f;

    // B-fragments of U: B[K=tau][N=b] = u[b, l0+tau, h]; 16 K-slices of 4.
    // Single EXEC toggle for the whole padded-column block.
    v2f ub[16];
    #pragma unroll
    for (int s = 0; s < 16; ++s) { ub[s].x = 0.0f; ub[s].y = 0.0f; }
    if (nl < B_) {
        const size_t base = ((size_t)nl * L_ + l0) * H_ + h;
        #pragma unroll
        for (int s = 0; s < 16; ++s) {
            const int t0 = 4 * s + 2 * half;
            ub[s].x = u[base + (size_t)t0 * H_];
            ub[s].y = u[base + (size_t)(t0 + 1) * H_];
        }
    }
    __syncthreads();

    const float Dh = Dvec[h];

    // ---- intra-chunk Toeplitz matmul, fused with skip connection ----
    #pragma unroll
    for (int mt = 0; mt < 4; ++mt) {
        const int tA = 16 * mt + nl;                 // A-matrix row (time t)
        v2f af[16];
        #pragma unroll
        for (int s = 0; s < 16; ++s) {               // preload A-frags (no waits in MMA chain)
            const int ridx = 63 - tA + 4 * s + 2 * half;  // in [0,126]
            af[s].x = rp[ridx];
            af[s].y = rp[ridx + 1];
        }
        v8f acc = {0.f,0.f,0.f,0.f,0.f,0.f,0.f,0.f};
        #pragma unroll
        for (int s = 0; s < 16; ++s)
            acc = wmma4(af[s], ub[s], acc);
        if (nl < B_) {
            #pragma unroll
            for (int r = 0; r < 8; ++r) {
                const int t = 16 * mt + r + 8 * half;    // C/D row
                const size_t idx = ((size_t)nl * L_ + (l0 + t)) * H_ + h;
                out[idx] = acc[r] + Dh * u[idx];
            }
        }
    }

    // ---- chunk state summaries: P = V * U (complex via 2 real matmuls) ----
    const float* Vtab[2] = { Vre, Vim };
    #pragma unroll
    for (int c = 0; c < 2; ++c) {
        const float* V = Vtab[c] + (size_t)h * N_ * Q_;
        #pragma unroll
        for (int mt = 0; mt < 4; ++mt) {
            // row n = 16*mt + nl; vectorized b64 fragment loads (tx even)
            const v2f* Vrow2 = reinterpret_cast<const v2f*>(V + (size_t)(16 * mt + nl) * Q_);
            v2f af[16];
            #pragma unroll
            for (int s = 0; s < 16; ++s)
                af[s] = Vrow2[(4 * s + 2 * half) >> 1];
            v8f acc = {0.f,0.f,0.f,0.f,0.f,0.f,0.f,0.f};
            #pragma unroll
            for (int s = 0; s < 16; ++s)
                acc = wmma4(af[s], ub[s], acc);
            if (nl < B_) {
                #pragma unroll
                for (int r = 0; r < 8; ++r) {
                    const int n = 16 * mt + r + 8 * half;
                    P[((((size_t)h * NCH + j) * 2 + c) * N_ + n) * B_ + nl] = acc[r];
                }
            }
        }
    }
}

// ---------------------------------------------------------------------------
// Kernel C: sequential scan over the 32 chunks (parallel over h,n,b):
//   carry_in[0] = 0;  P[j] <- carry_in[j];  carry = lamQ*carry + P_old[j]
// After this kernel, P[h][j] holds the complex carry-in state for chunk j.
// ---------------------------------------------------------------------------
__global__ __launch_bounds__(256) void s4d_scan(
    const float* __restrict__ lamQ, float* __restrict__ P)
{
    const int tid = blockIdx.x * blockDim.x + threadIdx.x;  // H_*N_*B_ threads
    const int h = tid / (N_ * B_);
    const int rem = tid % (N_ * B_);
    const int n = rem / B_, b = rem % B_;
    const float ar = lamQ[2 * (h * N_ + n)];
    const float ai = lamQ[2 * (h * N_ + n) + 1];
    float cr = 0.0f, ci = 0.0f;
    for (int jj = 0; jj < NCH; ++jj) {
        const size_t base = (((size_t)h * NCH + jj) * 2) * (N_ * B_)
                          + (size_t)n * B_ + b;
        const float prr = P[base];
        const float pii = P[base + N_ * B_];
        P[base]           = cr;
        P[base + N_ * B_] = ci;
        const float nr = ar * cr - ai * ci + prr;
        const float ni = ar * ci + ai * cr + pii;
        cr = nr; ci = ni;
    }
}

// ---------------------------------------------------------------------------
// Kernel D: inter-chunk contribution, one wave per (chunk j, head h):
//   Y2[t,b] = Gre[t,:]·Sre[:,b] + (-Gim)[t,:]·Sim[:,b];  out += Y2
// ---------------------------------------------------------------------------
__global__ __launch_bounds__(32) void s4d_inter(
    const float* __restrict__ Gre, const float* __restrict__ Gimn,
    const float* __restrict__ P, float* __restrict__ out)
{
    const int j = blockIdx.x, h = blockIdx.y;
    const int lane = threadIdx.x;
    const int half = lane >> 4;
    const int nl   = lane & 15;

    const float* Sre = P + (((size_t)h * NCH + j) * 2 + 0) * (N_ * B_);
    const float* Sim = Sre + N_ * B_;

    // B-fragments of carry-in states: B[K=n][N=b]; single EXEC toggle
    v2f sr[16], si[16];
    #pragma unroll
    for (int s = 0; s < 16; ++s) {
        sr[s].x = 0.f; sr[s].y = 0.f;
        si[s].x = 0.f; si[s].y = 0.f;
    }
    if (nl < B_) {
        #pragma unroll
        for (int s = 0; s < 16; ++s) {
            const int n0 = 4 * s + 2 * half;
            sr[s].x = Sre[(size_t)n0 * B_ + nl];
            sr[s].y = Sre[(size_t)(n0 + 1) * B_ + nl];
            si[s].x = Sim[(size_t)n0 * B_ + nl];
            si[s].y = Sim[(size_t)(n0 + 1) * B_ + nl];
        }
    }

    const float* Gr = Gre  + (size_t)h * Q_ * N_;
    const float* Gi = Gimn + (size_t)h * Q_ * N_;
    #pragma unroll
    for (int mt = 0; mt < 4; ++mt) {
        const v2f* Grow_r2 = reinterpret_cast<const v2f*>(Gr + (size_t)(16 * mt + nl) * N_);
        const v2f* Grow_i2 = reinterpret_cast<const v2f*>(Gi + (size_t)(16 * mt + nl) * N_);
        v2f gfr[16], gfi[16];
        #pragma unroll
        for (int s = 0; s < 16; ++s) {               // preload A-frags (b64 loads)
            const int e = (4 * s + 2 * half) >> 1;
            gfr[s] = Grow_r2[e];
            gfi[s] = Grow_i2[e];
        }
        v8f acc = {0.f,0.f,0.f,0.f,0.f,0.f,0.f,0.f};
        #pragma unroll
        for (int s = 0; s < 16; ++s)
            acc = wmma4(gfr[s], sr[s], acc);
        #pragma unroll
        for (int s = 0; s < 16; ++s)
            acc = wmma4(gfi[s], si[s], acc);
        if (nl < B_) {
            #pragma unroll
            for (int r = 0; r < 8; ++r) {
                const int t = 16 * mt + r + 8 * half;
                const size_t idx = ((size_t)nl * L_ + (j * Q_ + t)) * H_ + h;
                out[idx] += acc[r];
            }
        }
    }
}

// ---------------------------------------------------------------------------
extern "C" void kernel_launch(void* const* d_in, const int* in_sizes, int n_in,
                              void* d_out, int out_size, void* d_ws, size_t ws_size,
                              hipStream_t stream)
{
    (void)in_sizes; (void)n_in; (void)out_size; (void)ws_size;
    const float* u    = (const float*)d_in[0];
    const float* w_re = (const float*)d_in[1];
    const float* w_im = (const float*)d_in[2];
    const float* C_re = (const float*)d_in[3];
    const float* C_im = (const float*)d_in[4];
    const float* Dv   = (const float*)d_in[5];
    const float* dt   = (const float*)d_in[6];
    float* out = (float*)d_out;
    float* ws  = (float*)d_ws;

    size_t off = 0;
    float* Vre  = ws + off; off += (size_t)H_ * N_ * Q_;           // 8 MB
    float* Vim  = ws + off; off += (size_t)H_ * N_ * Q_;           // 8 MB
    float* Gre  = ws + off; off += (size_t)H_ * Q_ * N_;           // 8 MB
    float* Gimn = ws + off; off += (size_t)H_ * Q_ * N_;           // 8 MB
    float* kv   = ws + off; off += (size_t)H_ * Q_;                // 128 KB
    float* lamQ = ws + off; off += (size_t)H_ * N_ * 2;            // 256 KB
    float* P    = ws + off; off += (size_t)H_ * NCH * 2 * N_ * B_; // 67 MB

    s4d_setup<<<H_, 64, 0, stream>>>(w_re, w_im, C_re, C_im, dt,
                                     Vre, Vim, Gre, Gimn, kv, lamQ);
    s4d_chunk<<<dim3(NCH, H_), 32, 0, stream>>>(u, Dv, Vre, Vim, kv, P, out);
    s4d_scan<<<(H_ * N_ * B_) / 256, 256, 0, stream>>>(lamQ, P);
    s4d_inter<<<dim3(NCH, H_), 32, 0, stream>>>(Gre, Gimn, P, out);
}